// Hadmar_55594056679776
// MI455X (gfx1250) — compile-verified
//
#include <hip/hip_runtime.h>

// f32 WMMA fragment types for V_WMMA_F32_16X16X4_F32:
//  A (16x4 f32)  = 2 VGPRs -> float2 ext vector
//  B (4x16 f32)  = 2 VGPRs -> float2 ext vector
//  C/D (16x16)   = 8 VGPRs -> float8 ext vector
typedef __attribute__((ext_vector_type(2))) float v2f;
typedef __attribute__((ext_vector_type(8))) float v8f;

#define WPB 8   // waves per 256-thread block (wave32)
#define TPW 4   // tiles processed per wave (amortize H synthesis, allow prefetch)

// out = H * X * H for 16x16 fp32 tiles, H = Sylvester-Hadamard order 16.
// One wave handles one tile at a time: 8 chained V_WMMA_F32_16X16X4_F32.
__global__ __launch_bounds__(256) void hadamard16_wmma(
    const float* __restrict__ x, float* __restrict__ out, int ntiles) {

  const int lane = threadIdx.x & 31;
  const int r    = lane & 15;       // row (A/D) or column (B/D) index
  const bool low = lane < 16;       // low half-wave?

  // ---- Synthesize H fragments in-register (H symmetric => A-frag == B-frag).
  // A-layout chunk k (K = 4k..4k+3):
  //   low  lane L : v0 = H[L][4k+0], v1 = H[L][4k+1]
  //   high lane L : v0 = H[L][4k+2], v1 = H[L][4k+3]
  // B-layout chunk k wants H[4k+kk][n] at the same slots; H[i][j]==H[j][i].
  v2f h[4];
#pragma unroll
  for (int k = 0; k < 4; ++k) {
    const int c0 = 4 * k + (low ? 0 : 2);
    h[k].x = (__popc(r & c0) & 1) ? -1.0f : 1.0f;
    h[k].y = (__popc(r & (c0 + 1)) & 1) ? -1.0f : 1.0f;
  }

  const long wid   = (long)blockIdx.x * WPB + (threadIdx.x >> 5);
  const long tile0 = wid * TPW;
  const float* xp  = x + tile0 * 256;
  float* op        = out + tile0 * 256;

#pragma unroll
  for (int t = 0; t < TPW; ++t) {
    if (tile0 + t >= ntiles) break;        // wave-uniform: EXEC stays all-ones
    const float* tb = xp + (long)t * 256;

    if (t + 1 < TPW)                       // pull next 1KB tile toward us
      __builtin_prefetch(tb + 256, 0, 1);  // -> global_prefetch_b8

    // ---- Load X directly into A-fragment layout (global_load_b64 each):
    //   low  lane r : X[r][4k+0..1];  high lane r : X[r][4k+2..3]
    v2f xa[4];
#pragma unroll
    for (int k = 0; k < 4; ++k) {
      const int off = r * 16 + 4 * k + (low ? 0 : 2);
      xa[k] = *reinterpret_cast<const v2f*>(tb + off);
    }

    // ---- W = X * H   (4 chained K=4 WMMAs, f32 accumulate)
    v8f w = {0.f, 0.f, 0.f, 0.f, 0.f, 0.f, 0.f, 0.f};
#pragma unroll
    for (int k = 0; k < 4; ++k)
      w = __builtin_amdgcn_wmma_f32_16x16x4_f32(false, xa[k], false, h[k],
                                                (short)0, w, false, false);

    // ---- Re-layout W (C/D layout) into B fragments for the second multiply.
    // D layout: low lane n: w[i] = W[i][n]; high lane n: w[i] = W[8+i][n].
    // B chunk k needs: low lane: W[4k][n], W[4k+1][n]; high: W[4k+2][n], W[4k+3][n].
    // Only a half-wave swap (lane ^ 16) is required.
    float ws[8];
#pragma unroll
    for (int i = 0; i < 8; ++i) ws[i] = __shfl_xor(w[i], 16, 32);

    v2f wb[4];
    wb[0].x = low ? w[0]  : ws[2];  wb[0].y = low ? w[1]  : ws[3];
    wb[1].x = low ? w[4]  : ws[6];  wb[1].y = low ? w[5]  : ws[7];
    wb[2].x = low ? ws[0] : w[2];   wb[2].y = low ? ws[1] : w[3];
    wb[3].x = low ? ws[4] : w[6];   wb[3].y = low ? ws[5] : w[7];

    // ---- Z = H * W
    v8f z = {0.f, 0.f, 0.f, 0.f, 0.f, 0.f, 0.f, 0.f};
#pragma unroll
    for (int k = 0; k < 4; ++k)
      z = __builtin_amdgcn_wmma_f32_16x16x4_f32(false, h[k], false, wb[k],
                                                (short)0, z, false, false);

    // ---- Store D layout: low lane n holds Z[0..7][n], high lane Z[8..15][n].
    // Each store instruction writes two dense 64B runs (rows i and i+8).
    // Output is write-once -> non-temporal hint, keep L2 for the read stream.
    float* ob = op + (long)t * 256;
#pragma unroll
    for (int i = 0; i < 8; ++i) {
      const int row = (low ? 0 : 8) + i;
      __builtin_nontemporal_store(z[i], ob + row * 16 + r);
    }
  }
}

extern "C" void kernel_launch(void* const* d_in, const int* in_sizes, int n_in,
                              void* d_out, int out_size, void* d_ws, size_t ws_size,
                              hipStream_t stream) {
  const float* x = (const float*)d_in[0];   // [B, C, 16, 16] fp32
  // d_in[1] (hadmar) is unused: H is synthesized exactly in-register.
  float* out = (float*)d_out;

  const int ntiles = in_sizes[0] / 256;             // 262144 tiles of 16x16
  const int tiles_per_block = WPB * TPW;            // 32
  const int grid = (ntiles + tiles_per_block - 1) / tiles_per_block;

  hipLaunchKernelGGL(hadamard16_wmma, dim3(grid), dim3(256), 0, stream,
                     x, out, ntiles);
}